// multi_level_expert_64544768524769
// MI455X (gfx1250) — compile-verified
//
#include <hip/hip_runtime.h>
#include <math.h>

// ---------------- problem constants ----------------
#define BSZ     2
#define IN_N    784
#define O1_N    10816
#define O2_N    332928
#define FC1_K   665856          // O2_N * BSZ
#define FC1_OUT 128

#define FC1_KSLAB 2048
#define FC1_NBLK  326           // ceil(665856 / 2048)
#define LSE_NBLK  326

typedef __attribute__((ext_vector_type(2))) float v2f;
typedef __attribute__((ext_vector_type(8))) float v8f;

// ---------------- workspace layout (floats) ----------------
// flat (y2, b-major)        : FC1_K
// y1                        : 2*O1_N
// cg0 (coef[2][4], gate[4]) : 12
// cg1                       : 12
// fc1 partials              : FC1_NBLK*128
// h                         : 128
// lse partials (m,s pairs)  : 2*LSE_NBLK
// lse                       : 1
static constexpr size_t OFF_FLAT = 0;
static constexpr size_t OFF_Y1   = OFF_FLAT + FC1_K;
static constexpr size_t OFF_CG0  = OFF_Y1 + 2 * O1_N;
static constexpr size_t OFF_CG1  = OFF_CG0 + 12;
static constexpr size_t OFF_PART = OFF_CG1 + 12;
static constexpr size_t OFF_H    = OFF_PART + (size_t)FC1_NBLK * 128;
static constexpr size_t OFF_LP   = OFF_H + 128;
static constexpr size_t OFF_LSE  = OFF_LP + 2 * LSE_NBLK;

// ---------------- K1/K3: gating + tiny hidden layer ----------------
// acc[0..3]  = logits[b][e]    (b major)
// acc[4..11] = pre[e][b][h] laid out as [b][e*2+h]
__global__ void __launch_bounds__(256)
moe_gate_kernel(const float* __restrict__ xin, int n,
                const float* __restrict__ wg,
                const float* __restrict__ w1,
                const float* __restrict__ b1,
                float* __restrict__ out12) {
  __shared__ float red[12][256];
  const int tid = threadIdx.x;
  float acc[12];
#pragma unroll
  for (int c = 0; c < 12; ++c) acc[c] = 0.f;

  for (int i = tid; i < n; i += 256) {
    float x0 = xin[i];
    float x1 = xin[n + i];
    float g0 = wg[i * 2 + 0], g1 = wg[i * 2 + 1];
    float w00 = w1[i * 2 + 0], w01 = w1[i * 2 + 1];            // e=0, h=0/1
    float w10 = w1[(size_t)n * 2 + i * 2 + 0];                 // e=1, h=0
    float w11 = w1[(size_t)n * 2 + i * 2 + 1];                 // e=1, h=1
    acc[0] += x0 * g0;  acc[1] += x0 * g1;
    acc[2] += x1 * g0;  acc[3] += x1 * g1;
    acc[4] += x0 * w00; acc[5] += x0 * w01; acc[6]  += x0 * w10; acc[7]  += x0 * w11;
    acc[8] += x1 * w00; acc[9] += x1 * w01; acc[10] += x1 * w10; acc[11] += x1 * w11;
  }
#pragma unroll
  for (int c = 0; c < 12; ++c) red[c][tid] = acc[c];
  __syncthreads();
  for (int s = 128; s > 0; s >>= 1) {
    if (tid < s) {
#pragma unroll
      for (int c = 0; c < 12; ++c) red[c][tid] += red[c][tid + s];
    }
    __syncthreads();
  }
  if (tid == 0) {
    for (int b = 0; b < 2; ++b) {
      float l0 = red[b * 2 + 0][0], l1 = red[b * 2 + 1][0];
      float m = fmaxf(l0, l1);
      float e0 = expf(l0 - m), e1 = expf(l1 - m);
      float inv = 1.f / (e0 + e1);
      float g0 = e0 * inv, g1 = e1 * inv;
      // coef[b][e*2+h] = gate[e] * relu(pre[e][b][h] + b1[e][h])
      out12[b * 4 + 0] = g0 * fmaxf(red[4 + b * 4 + 0][0] + b1[0], 0.f);
      out12[b * 4 + 1] = g0 * fmaxf(red[4 + b * 4 + 1][0] + b1[1], 0.f);
      out12[b * 4 + 2] = g1 * fmaxf(red[4 + b * 4 + 2][0] + b1[2], 0.f);
      out12[b * 4 + 3] = g1 * fmaxf(red[4 + b * 4 + 3][0] + b1[3], 0.f);
      out12[8 + b * 2 + 0] = g0;
      out12[8 + b * 2 + 1] = g1;
    }
  }
}

// ---------------- K2/K4: rank-4 expert combine ----------------
// out[b*O + o] = sum_{eh} cg[b*4+eh] * w2[eh*O + o] + sum_e cg[8+b*2+e]*b2[e*O+o]
__global__ void __launch_bounds__(256)
expert_out_kernel(const float* __restrict__ w2,
                  const float* __restrict__ b2,
                  const float* __restrict__ cg,
                  int O, float* __restrict__ out) {
  float c0 = cg[0], c1 = cg[1], c2 = cg[2], c3 = cg[3];
  float c4 = cg[4], c5 = cg[5], c6 = cg[6], c7 = cg[7];
  float ga0 = cg[8], ga1 = cg[9], gb0 = cg[10], gb1 = cg[11];
  for (int o = blockIdx.x * blockDim.x + threadIdx.x; o < O;
       o += gridDim.x * blockDim.x) {
    float w0 = w2[o];
    float w1 = w2[(size_t)O + o];
    float w2v = w2[(size_t)2 * O + o];
    float w3 = w2[(size_t)3 * O + o];
    float bb0 = b2[o];
    float bb1 = b2[(size_t)O + o];
    out[o]             = c0 * w0 + c1 * w1 + c2 * w2v + c3 * w3 + ga0 * bb0 + ga1 * bb1;
    out[(size_t)O + o] = c4 * w0 + c5 * w1 + c6 * w2v + c7 * w3 + gb0 * bb0 + gb1 * bb1;
  }
}

// ---------------- K5: fc1 GEMV slabs via V_WMMA_F32_16X16X4_F32 ----------------
// Wave w handles output rows j0=16w..16w+15 over this block's K-slab.
// A[m][k] = W[j0+m][kb+k]; B[k][n] = flat[kb+k] (replicated over N).
// A layout (ISA 16x4 f32): lane L -> row j0+(L&15), k-pair 2*(L>>4).
// B layout matches same per-lane k indexing.
__global__ void __launch_bounds__(256)
fc1_wmma_kernel(const float* __restrict__ W,
                const float* __restrict__ flat,
                float* __restrict__ partial) {
  const int wave = threadIdx.x >> 5;
  const int lane = threadIdx.x & 31;
  const int j = wave * 16 + (lane & 15);
  const int koff = 2 * (lane >> 4);
  const size_t rowbase = (size_t)j * FC1_K;

  int kb = blockIdx.x * FC1_KSLAB;
  int kend = kb + FC1_KSLAB;
  if (kend > FC1_K) kend = FC1_K;

  v8f c = {0.f, 0.f, 0.f, 0.f, 0.f, 0.f, 0.f, 0.f};
#pragma unroll 4
  for (; kb < kend; kb += 4) {
    v2f a = *(const v2f*)(W + rowbase + kb + koff);
    v2f b = *(const v2f*)(flat + kb + koff);
    c = __builtin_amdgcn_wmma_f32_16x16x4_f32(false, a, false, b,
                                              (short)0, c, false, false);
  }
  if ((lane & 15) == 0) {
    // lane 0 -> rows 0..7 of the tile, lane 16 -> rows 8..15 (N==0 column)
    float* p = partial + (size_t)blockIdx.x * 128 + wave * 16 + (lane >> 4) * 8;
#pragma unroll
    for (int i = 0; i < 8; ++i) p[i] = c[i];
  }
}

// ---------------- K6: reduce fc1 partials + bias + relu ----------------
__global__ void fc1_finish_kernel(const float* __restrict__ partial,
                                  const float* __restrict__ bias,
                                  float* __restrict__ h) {
  int j = threadIdx.x;  // 128 threads
  float s = 0.f;
  for (int b = 0; b < FC1_NBLK; ++b) s += partial[(size_t)b * 128 + j];
  h[j] = fmaxf(s + bias[j], 0.f);
}

// ---------------- K7: fc2 GEMV via WMMA (16 output rows/wave, K=128) --------
__global__ void __launch_bounds__(256)
fc2_wmma_kernel(const float* __restrict__ W,
                const float* __restrict__ h,
                const float* __restrict__ bias,
                float* __restrict__ logits) {
  const int wave = threadIdx.x >> 5;
  const int lane = threadIdx.x & 31;
  const int o0 = (blockIdx.x * 8 + wave) * 16;
  const int o = o0 + (lane & 15);
  const int koff = 2 * (lane >> 4);
  const float* row = W + (size_t)o * 128;

  v8f c = {0.f, 0.f, 0.f, 0.f, 0.f, 0.f, 0.f, 0.f};
#pragma unroll 8
  for (int kb = 0; kb < 128; kb += 4) {
    v2f a = *(const v2f*)(row + kb + koff);
    v2f b = *(const v2f*)(h + kb + koff);
    c = __builtin_amdgcn_wmma_f32_16x16x4_f32(false, a, false, b,
                                              (short)0, c, false, false);
  }
  if ((lane & 15) == 0) {
    int ob = o0 + (lane >> 4) * 8;
#pragma unroll
    for (int i = 0; i < 8; ++i) logits[ob + i] = c[i] + bias[ob + i];
  }
}

// ---------------- K8: per-block online (max, sumexp) ----------------
__global__ void __launch_bounds__(256)
lse_part_kernel(const float* __restrict__ x, int n, float* __restrict__ part) {
  __shared__ float sm[256], ss[256];
  const int tid = threadIdx.x;
  float m = -INFINITY, s = 0.f;
  for (int i = blockIdx.x * blockDim.x + tid; i < n; i += gridDim.x * blockDim.x) {
    float v = x[i];
    if (v > m) { s = s * expf(m - v) + 1.f; m = v; }
    else       { s += expf(v - m); }
  }
  sm[tid] = m; ss[tid] = s;
  __syncthreads();
  for (int st = 128; st > 0; st >>= 1) {
    if (tid < st) {
      float m2 = sm[tid + st], s2 = ss[tid + st];
      float M = fmaxf(sm[tid], m2);
      float S = (M == -INFINITY) ? 0.f
                                 : ss[tid] * expf(sm[tid] - M) + s2 * expf(m2 - M);
      sm[tid] = M; ss[tid] = S;
    }
    __syncthreads();
  }
  if (tid == 0) { part[blockIdx.x * 2] = sm[0]; part[blockIdx.x * 2 + 1] = ss[0]; }
}

// ---------------- K9: merge block pairs -> logsumexp ----------------
__global__ void lse_final_kernel(const float* __restrict__ part, int nb,
                                 float* __restrict__ lse) {
  if (threadIdx.x == 0 && blockIdx.x == 0) {
    float m = -INFINITY, s = 0.f;
    for (int i = 0; i < nb; ++i) {
      float m2 = part[2 * i], s2 = part[2 * i + 1];
      float M = fmaxf(m, m2);
      float S = (M == -INFINITY) ? 0.f : s * expf(m - M) + s2 * expf(m2 - M);
      m = M; s = S;
    }
    *lse = m + logf(s);
  }
}

// ---------------- K10: subtract lse in place ----------------
__global__ void __launch_bounds__(256)
sub_lse_kernel(float* __restrict__ x, int n, const float* __restrict__ lse) {
  float L = *lse;
  for (int i = blockIdx.x * blockDim.x + threadIdx.x; i < n;
       i += gridDim.x * blockDim.x) {
    x[i] -= L;
  }
}

// ---------------- host-side orchestration ----------------
extern "C" void kernel_launch(void* const* d_in, const int* in_sizes, int n_in,
                              void* d_out, int out_size, void* d_ws, size_t ws_size,
                              hipStream_t stream) {
  const float* x     = (const float*)d_in[0];
  const float* wg0   = (const float*)d_in[1];
  const float* w1_0  = (const float*)d_in[2];
  const float* b1_0  = (const float*)d_in[3];
  const float* w2_0  = (const float*)d_in[4];
  const float* b2_0  = (const float*)d_in[5];
  const float* wg1   = (const float*)d_in[6];
  const float* w1_1  = (const float*)d_in[7];
  const float* b1_1  = (const float*)d_in[8];
  const float* w2_1  = (const float*)d_in[9];
  const float* b2_1  = (const float*)d_in[10];
  const float* fc1_w = (const float*)d_in[11];
  const float* fc1_b = (const float*)d_in[12];
  const float* fc2_w = (const float*)d_in[13];
  const float* fc2_b = (const float*)d_in[14];
  float* out = (float*)d_out;
  float* ws  = (float*)d_ws;

  // MoE layer 0: gates/coefs then y1 [2, O1]
  moe_gate_kernel<<<1, 256, 0, stream>>>(x, IN_N, wg0, w1_0, b1_0, ws + OFF_CG0);
  expert_out_kernel<<<(O1_N + 255) / 256, 256, 0, stream>>>(
      w2_0, b2_0, ws + OFF_CG0, O1_N, ws + OFF_Y1);

  // MoE layer 1: gates/coefs then flat = y2 [2, O2]
  moe_gate_kernel<<<1, 256, 0, stream>>>(ws + OFF_Y1, O1_N, wg1, w1_1, b1_1,
                                         ws + OFF_CG1);
  expert_out_kernel<<<(O2_N + 255) / 256, 256, 0, stream>>>(
      w2_1, b2_1, ws + OFF_CG1, O2_N, ws + OFF_FLAT);

  // fc1: 341MB streamed through WMMA, split-K partials then reduce
  fc1_wmma_kernel<<<FC1_NBLK, 256, 0, stream>>>(fc1_w, ws + OFF_FLAT,
                                                ws + OFF_PART);
  fc1_finish_kernel<<<1, 128, 0, stream>>>(ws + OFF_PART, fc1_b, ws + OFF_H);

  // fc2: 170MB streamed through WMMA -> raw logits in d_out
  fc2_wmma_kernel<<<O2_N / (16 * 8), 256, 0, stream>>>(fc2_w, ws + OFF_H, fc2_b,
                                                       out);

  // log_softmax over all 332928 logits
  lse_part_kernel<<<LSE_NBLK, 256, 0, stream>>>(out, O2_N, ws + OFF_LP);
  lse_final_kernel<<<1, 32, 0, stream>>>(ws + OFF_LP, LSE_NBLK, ws + OFF_LSE);
  sub_lse_kernel<<<(O2_N + 255) / 256, 256, 0, stream>>>(out, O2_N,
                                                         ws + OFF_LSE);
}